// Generator_90683939488402
// MI455X (gfx1250) — compile-verified
//
#include <hip/hip_runtime.h>
#include <math.h>

typedef __attribute__((ext_vector_type(2))) float v2f;
typedef __attribute__((ext_vector_type(8))) float v8f;

#define BROWS 131072
#define WPB   4            // waves per block (wave32)
#define TPB   (WPB * 32)
#define RBLK  1024         // grid.y row-tile blocks (B/16 = 8192 tiles total)
#define NEGV  -1000000000.0f

static __device__ __forceinline__ v8f wmma4(v2f a, v2f b, v8f c) {
  return __builtin_amdgcn_wmma_f32_16x16x4_f32(
      /*neg_a=*/false, a, /*neg_b=*/false, b,
      /*c_mod=*/(short)0, c, /*reuse_a=*/false, /*reuse_b=*/false);
}

// ---------------------------------------------------------------- zero stats
__global__ void k_zero(float* p, int n) {
  int i = blockIdx.x * blockDim.x + threadIdx.x;
  if (i < n) p[i] = 0.f;
}

// ------------------------------------------------ pad W4 [128,25] -> [128,32]
__global__ void k_pad(const float* __restrict__ W4, const float* __restrict__ b4,
                      float* __restrict__ Wp, float* __restrict__ bp) {
  int i = blockIdx.x * blockDim.x + threadIdx.x;   // 0 .. 4095
  if (i < 128 * 32) {
    int r = i >> 5, c = i & 31;
    Wp[i] = (c < 25) ? W4[r * 25 + c] : 0.f;
  }
  if (i < 32) bp[i] = (i < 25) ? b4[i] : 0.f;
}

// ---------------------------------------------------------------- fused GEMM
// Y = act(A) @ W + bias;  act(a) = lrelu(a*s[k]+t[k]) when NORM.
// Each wave owns a 16x32 output tile (two 16x16 WMMA accumulators), K stepped
// by 4 via V_WMMA_F32_16X16X4_F32.  Fully unrolled K loop with base-pointer +
// compile-time immediate offsets -> no per-step address arithmetic.
template <int K, int N, bool NORM, bool STATS>
__global__ __launch_bounds__(TPB) void k_gemm(
    const float* __restrict__ A, const float* __restrict__ W,
    const float* __restrict__ bias,
    const float* __restrict__ sPrev, const float* __restrict__ tPrev,
    float* __restrict__ Y, float* __restrict__ stSum, float* __restrict__ stSq,
    int M)
{
  const int lane = threadIdx.x & 31;
  const int wave = threadIdx.x >> 5;
  const int col0 = (blockIdx.x * WPB + wave) << 5;   // 32 columns per wave
  if (col0 >= N) return;                              // wave-uniform

  const int n0 = col0 + (lane & 15);
  const int khalf = (lane >> 4) << 1;                 // 0 or 2 (K sub-lane)
  const float bias0 = bias[n0];
  const float bias1 = bias[n0 + 16];

  // per-wave base pointers; all hot-loop offsets are compile-time immediates
  const float* Wb = W + (size_t)khalf * N + n0;       // +16 for second tile
  const float* sb = NORM ? (sPrev + khalf) : nullptr;
  const float* tb = NORM ? (tPrev + khalf) : nullptr;

  float sum0 = 0.f, sq0 = 0.f, sum1 = 0.f, sq1 = 0.f;
  const int mtiles = M >> 4;
  constexpr int KMAIN = (K / 4) * 4;

  for (int mt = blockIdx.y; mt < mtiles; mt += gridDim.y) {
    const int row = (mt << 4) + (lane & 15);          // A operand: lane = M
    const float* Aro = A + (size_t)row * K + khalf;
    v8f c0 = {}, c1 = {};

#pragma unroll
    for (int k0 = 0; k0 < KMAIN; k0 += 4) {
      v2f a = *(const v2f*)(Aro + k0);                // b64, immediate offset
      if (NORM) {                                     // fused BN + leaky-relu
        v2f s = *(const v2f*)(sb + k0);
        v2f t = *(const v2f*)(tb + k0);
        float x0 = fmaf(a[0], s[0], t[0]);
        float x1 = fmaf(a[1], s[1], t[1]);
        a[0] = x0 > 0.f ? x0 : 0.2f * x0;
        a[1] = x1 > 0.f ? x1 : 0.2f * x1;
      }
      v2f b0, b1;
      b0[0] = Wb[k0 * N];                             // immediate offsets
      b0[1] = Wb[(k0 + 1) * N];
      b1[0] = Wb[k0 * N + 16];
      b1[1] = Wb[(k0 + 1) * N + 16];
      c0 = wmma4(a, b0, c0);
      c1 = wmma4(a, b1, c1);
    }

    if constexpr (K % 4 != 0) {
      // remainder (K=50): only khalf==0 lanes carry k=KMAIN..KMAIN+1.
      // Clamped offsets + value selects -> no exec divergence.
      const bool lo = (khalf == 0);
      const int ko = lo ? KMAIN : 0;
      v2f a = *(const v2f*)(Aro + ko);
      if (NORM) {
        v2f s = *(const v2f*)(sb + ko);
        v2f t = *(const v2f*)(tb + ko);
        float x0 = fmaf(a[0], s[0], t[0]);
        float x1 = fmaf(a[1], s[1], t[1]);
        a[0] = x0 > 0.f ? x0 : 0.2f * x0;
        a[1] = x1 > 0.f ? x1 : 0.2f * x1;
      }
      a[0] = lo ? a[0] : 0.f;
      a[1] = lo ? a[1] : 0.f;
      v2f b0, b1;
      b0[0] = lo ? Wb[ko * N]            : 0.f;
      b0[1] = lo ? Wb[(ko + 1) * N]      : 0.f;
      b1[0] = lo ? Wb[ko * N + 16]       : 0.f;
      b1[1] = lo ? Wb[(ko + 1) * N + 16] : 0.f;
      c0 = wmma4(a, b0, c0);
      c1 = wmma4(a, b1, c1);
    }

    // C/D layout: VGPR r -> M = r (lanes 0-15) or 8+r (lanes 16-31)
    const int rbase = (mt << 4) + ((lane >> 4) << 3);
    float* Yp = Y + (size_t)rbase * N + n0;
#pragma unroll
    for (int r = 0; r < 8; ++r) {
      const float v0 = c0[r] + bias0;
      const float v1 = c1[r] + bias1;
      Yp[r * N]      = v0;                            // immediate offsets
      Yp[r * N + 16] = v1;
      if (STATS) {
        sum0 += v0; sq0 = fmaf(v0, v0, sq0);
        sum1 += v1; sq1 = fmaf(v1, v1, sq1);
      }
    }
  }

  if (STATS) {
    // lanes L and L^16 hold the same columns -> pair-reduce, then one atomic
    sum0 += __shfl_xor(sum0, 16, 32);
    sq0  += __shfl_xor(sq0,  16, 32);
    sum1 += __shfl_xor(sum1, 16, 32);
    sq1  += __shfl_xor(sq1,  16, 32);
    if (lane < 16) {
      unsafeAtomicAdd(&stSum[n0],      sum0);
      unsafeAtomicAdd(&stSq[n0],       sq0);
      unsafeAtomicAdd(&stSum[n0 + 16], sum1);
      unsafeAtomicAdd(&stSq[n0 + 16],  sq1);
    }
  }
}

// ------------------------------------------------- BN stats -> affine (s, t)
__global__ void k_fin(const float* __restrict__ sum, const float* __restrict__ sq,
                      const float* __restrict__ g,   const float* __restrict__ be,
                      float* __restrict__ s, float* __restrict__ t, int n)
{
  int i = blockIdx.x * blockDim.x + threadIdx.x;
  if (i >= n) return;
  const float invB = 1.f / (float)BROWS;
  const float mu   = sum[i] * invB;
  const float var  = sq[i] * invB - mu * mu;     // biased variance
  const float sc   = g[i] * rsqrtf(var + 1e-5f);
  s[i] = sc;
  t[i] = be[i] - mu * sc;
}

// --------------------------------------------- masked Gumbel top-K sampling
__global__ __launch_bounds__(256) void k_gumbel(
    const float* __restrict__ logits,   // [B, 32] padded
    const float* __restrict__ u,        // [15, B, 25]
    float* __restrict__ out, int Btot)
{
  int b = blockIdx.x * blockDim.x + threadIdx.x;
  if (b >= Btot) return;
  float lg[25];
#pragma unroll
  for (int c = 0; c < 25; ++c) lg[c] = logits[(size_t)b * 32 + c];
  unsigned mask = 0x1FFFFFFu;
#pragma unroll 1
  for (int k = 0; k < 15; ++k) {
    const float* uk = u + ((size_t)k * Btot + b) * 25;
    float best = -3.4e38f;
    int   bi   = 0;
#pragma unroll
    for (int c = 0; c < 25; ++c) {
      const float lv = ((mask >> c) & 1u) ? lg[c] : NEGV;
      const float nz = -__logf(-__logf(uk[c]));
      const float gv = (lv + nz) * 2.0f;     // /TEMP, TEMP=0.5 (monotone)
      if (gv > best) { best = gv; bi = c; }  // strict '>' == first-max rule
    }
    out[(size_t)b * 15 + k] = (float)bi * (1.0f / 24.0f);
    mask &= ~(1u << bi);
  }
}

// ---------------------------------------------------------------------------
extern "C" void kernel_launch(void* const* d_in, const int* in_sizes, int n_in,
                              void* d_out, int out_size, void* d_ws, size_t ws_size,
                              hipStream_t stream) {
  (void)in_sizes; (void)n_in; (void)out_size; (void)ws_size;
  const float* z   = (const float*)d_in[0];
  const float* u   = (const float*)d_in[1];
  const float* W1  = (const float*)d_in[2];
  const float* b1  = (const float*)d_in[3];
  const float* g1  = (const float*)d_in[4];
  const float* be1 = (const float*)d_in[5];
  const float* W2  = (const float*)d_in[6];
  const float* b2  = (const float*)d_in[7];
  const float* g2  = (const float*)d_in[8];
  const float* be2 = (const float*)d_in[9];
  const float* W3  = (const float*)d_in[10];
  const float* b3  = (const float*)d_in[11];
  const float* g3  = (const float*)d_in[12];
  const float* be3 = (const float*)d_in[13];
  const float* W4  = (const float*)d_in[14];
  const float* b4  = (const float*)d_in[15];
  float* out = (float*)d_out;

  float* ws   = (float*)d_ws;
  // stats (zeroed each launch)
  float* st1s = ws + 0;    float* st1q = ws + 128;
  float* st2s = ws + 256;  float* st2q = ws + 512;
  float* st3s = ws + 768;  float* st3q = ws + 896;
  // folded BN affine params
  float* s1 = ws + 1024;   float* t1 = ws + 1152;
  float* s2 = ws + 1280;   float* t2 = ws + 1536;
  float* s3 = ws + 1792;   float* t3 = ws + 1920;
  // padded layer-4 weights
  float* Wp = ws + 2048;   // 128*32 = 4096 floats
  float* bp = ws + 6144;   // 32 floats
  // activations (Y1 region reused for padded logits after Y1 is consumed)
  float* Y1     = ws + 8192;
  float* Y2     = Y1 + (size_t)BROWS * 128;
  float* Y3     = Y2 + (size_t)BROWS * 256;
  float* logits = Y1;                         // [B, 32]

  const int B = BROWS;

  k_zero<<<4, 256, 0, stream>>>(ws, 1024);
  k_pad<<<16, 256, 0, stream>>>(W4, b4, Wp, bp);

  // layer 1: [B,50] @ [50,128]  (raw input, no BN on A)
  k_gemm<50, 128, false, true><<<dim3(1, RBLK), TPB, 0, stream>>>(
      z, W1, b1, nullptr, nullptr, Y1, st1s, st1q, B);
  k_fin<<<1, 128, 0, stream>>>(st1s, st1q, g1, be1, s1, t1, 128);

  // layer 2: [B,128] @ [128,256]
  k_gemm<128, 256, true, true><<<dim3(2, RBLK), TPB, 0, stream>>>(
      Y1, W2, b2, s1, t1, Y2, st2s, st2q, B);
  k_fin<<<1, 256, 0, stream>>>(st2s, st2q, g2, be2, s2, t2, 256);

  // layer 3: [B,256] @ [256,128]
  k_gemm<256, 128, true, true><<<dim3(1, RBLK), TPB, 0, stream>>>(
      Y2, W3, b3, s2, t2, Y3, st3s, st3q, B);
  k_fin<<<1, 128, 0, stream>>>(st3s, st3q, g3, be3, s3, t3, 128);

  // layer 4: [B,128] @ [128,32] (padded) -> logits, single wave per block
  k_gemm<128, 32, true, false><<<dim3(1, RBLK), 32, 0, stream>>>(
      Y3, Wp, bp, s3, t3, logits, nullptr, nullptr, B);

  // Gumbel top-15 sampling
  k_gumbel<<<B / 256, 256, 0, stream>>>(logits, u, out, B);
}